// DecoderRNN_58978490908700
// MI455X (gfx1250) — compile-verified
//
#include <hip/hip_runtime.h>
#include <math.h>

#define HH   2048
#define G4   (4 * HH)
#define NSPL 4
#define KC   (HH / NSPL)   // 512

typedef float v2f __attribute__((ext_vector_type(2)));
typedef float v8f __attribute__((ext_vector_type(8)));

struct HPtrs { const float* p[7]; };

// 16-row GEMV tile over K range [kbeg, kend) using V_WMMA_F32_16X16X4_F32.
// W row-major with leading dim HH. sx = LDS copy of x, indexed from kbeg.
// Result: column 0 of the 16x16 f32 accumulator -> lanes 0 (M=0..7) / 16 (M=8..15).
__device__ __forceinline__ v8f gemv16_wmma(const float* __restrict__ W,
                                           const float* __restrict__ sx,
                                           int m0, int kbeg, int kend, int lane) {
    const int half = lane >> 4;          // 0: K=0,1 ; 1: K=2,3  (f32 A layout)
    const int rl   = lane & 15;          // row within tile
    const float* wrow = W + (size_t)(m0 + rl) * HH + kbeg + 2 * half;
    const float* xv   = sx + 2 * half;
    v8f acc = {0.f, 0.f, 0.f, 0.f, 0.f, 0.f, 0.f, 0.f};
    const v2f bz = {0.f, 0.f};
    const int kn = kend - kbeg;
#pragma unroll 4
    for (int k = 0; k < kn; k += 4) {
        v2f a = *(const v2f*)(wrow + k);           // global_load_b64, streams W rows
        v2f b;
        b.x = xv[k];                                // ds_load (broadcast per half-wave)
        b.y = xv[k + 1];
        b = (rl == 0) ? b : bz;                     // x only in B column 0 (lanes 0,16)
        acc = __builtin_amdgcn_wmma_f32_16x16x4_f32(
            false, a, false, b, (short)0, acc, false, false);
    }
    return acc;
}

// Phase 1 (parallel): z_layer = Whh_layer @ h_layer + bih + bhh  (+ Wih1 @ x for layer 0)
__global__ __launch_bounds__(256) void gemv_all_hh(
    const float* __restrict__ Whh1, const float* __restrict__ Whh,
    const float* __restrict__ bih1, const float* __restrict__ bhh1,
    const float* __restrict__ bih,  const float* __restrict__ bhh,
    const float* __restrict__ Wih1, const float* __restrict__ x,
    HPtrs hs, float* __restrict__ z)
{
    __shared__ float sx[HH];
    const int layer = blockIdx.y;
    const float* W  = (layer == 0) ? Whh1 : Whh + (size_t)(layer - 1) * G4 * HH;
    const float* bi = (layer == 0) ? bih1 : bih + (size_t)(layer - 1) * G4;
    const float* bh = (layer == 0) ? bhh1 : bhh + (size_t)(layer - 1) * G4;
    const float* hv = hs.p[layer];
    float* zl = z + (size_t)layer * G4;

    for (int i = threadIdx.x; i < HH; i += blockDim.x) sx[i] = hv[i];
    __syncthreads();

    const int wave = threadIdx.x >> 5, lane = threadIdx.x & 31;
    const int m0 = blockIdx.x * 128 + wave * 16;
    v8f acc = gemv16_wmma(W, sx, m0, 0, HH, lane);

    const int rl = lane & 15, half = lane >> 4;
    if (rl == 0) {
        float x0 = 0.f, x1 = 0.f;
        if (layer == 0) { x0 = x[0]; x1 = x[1]; }
        const int mb = m0 + 8 * half;
#pragma unroll
        for (int r = 0; r < 8; ++r) {
            const int m = mb + r;
            float y = acc[r] + bi[m] + bh[m];
            if (layer == 0) y += Wih1[2 * m] * x0 + Wih1[2 * m + 1] * x1;
            zl[m] = y;
        }
    }
}

// Phase 2 (sequential chain): split-K partials of Wih_k @ (xa [+ xb])
__global__ __launch_bounds__(256) void gemv_ih_split(
    const float* __restrict__ W, const float* __restrict__ xa,
    const float* __restrict__ xb, float* __restrict__ parts)
{
    __shared__ float sx[KC];
    const int split = blockIdx.y;
    const int kbeg  = split * KC;
    for (int i = threadIdx.x; i < KC; i += blockDim.x) {
        float v = xa[kbeg + i];
        if (xb) v += xb[kbeg + i];
        sx[i] = v;
    }
    __syncthreads();

    const int wave = threadIdx.x >> 5, lane = threadIdx.x & 31;
    const int m0 = blockIdx.x * 128 + wave * 16;
    v8f acc = gemv16_wmma(W, sx, m0, kbeg, kbeg + KC, lane);

    const int rl = lane & 15, half = lane >> 4;
    if (rl == 0) {
        float* po = parts + (size_t)split * G4;
        const int mb = m0 + 8 * half;
#pragma unroll
        for (int r = 0; r < 8; ++r) po[mb + r] = acc[r];
    }
}

// LSTM cell pointwise: gates i,f,g,o at rows j, H+j, 2H+j, 3H+j (PyTorch order)
__global__ __launch_bounds__(256) void lstm_cell(
    const float* __restrict__ z, const float* __restrict__ parts,
    const float* __restrict__ cin, float* __restrict__ hout, float* __restrict__ cout)
{
    const int j = blockIdx.x * blockDim.x + threadIdx.x;
    if (j >= HH) return;
    float gi = z[j], gf = z[HH + j], gg = z[2 * HH + j], go = z[3 * HH + j];
    if (parts) {
#pragma unroll
        for (int s = 0; s < NSPL; ++s) {         // fixed-order sum -> deterministic
            const float* p = parts + (size_t)s * G4;
            gi += p[j]; gf += p[HH + j]; gg += p[2 * HH + j]; go += p[3 * HH + j];
        }
    }
    const float si = 1.f / (1.f + expf(-gi));
    const float sf = 1.f / (1.f + expf(-gf));
    const float so = 1.f / (1.f + expf(-go));
    const float c2 = sf * cin[j] + si * tanhf(gg);
    hout[j] = so * tanhf(c2);
    cout[j] = c2;
}

// probs = softmax(Wout @ h + bout), OUT = 2
__global__ __launch_bounds__(256) void proj_softmax(
    const float* __restrict__ Wout, const float* __restrict__ bout,
    const float* __restrict__ h, float* __restrict__ probs)
{
    __shared__ float s0[256], s1[256];
    float a0 = 0.f, a1 = 0.f;
    for (int i = threadIdx.x; i < HH; i += 256) {
        const float hv = h[i];
        a0 += Wout[i] * hv;
        a1 += Wout[HH + i] * hv;
    }
    s0[threadIdx.x] = a0; s1[threadIdx.x] = a1;
    __syncthreads();
    for (int s = 128; s > 0; s >>= 1) {
        if (threadIdx.x < s) {
            s0[threadIdx.x] += s0[threadIdx.x + s];
            s1[threadIdx.x] += s1[threadIdx.x + s];
        }
        __syncthreads();
    }
    if (threadIdx.x == 0) {
        const float l0 = s0[0] + bout[0], l1 = s1[0] + bout[1];
        const float m = fmaxf(l0, l1);
        const float e0 = expf(l0 - m), e1 = expf(l1 - m);
        const float inv = 1.f / (e0 + e1);
        probs[0] = e0 * inv;
        probs[1] = e1 * inv;
    }
}

extern "C" void kernel_launch(void* const* d_in, const int* in_sizes, int n_in,
                              void* d_out, int out_size, void* d_ws, size_t ws_size,
                              hipStream_t stream) {
    // setup_inputs() order: x, (h1,c1)...(h7,c7), Wih1, Whh1, bih1, bhh1,
    //                       Wih[6], Whh[6], bih[6], bhh[6], Wout, bout
    const float* x = (const float*)d_in[0];
    const float* hin[7]; const float* cin[7];
    for (int i = 0; i < 7; ++i) {
        hin[i] = (const float*)d_in[1 + 2 * i];
        cin[i] = (const float*)d_in[2 + 2 * i];
    }
    const float* Wih1 = (const float*)d_in[15];
    const float* Whh1 = (const float*)d_in[16];
    const float* bih1 = (const float*)d_in[17];
    const float* bhh1 = (const float*)d_in[18];
    const float* Wih  = (const float*)d_in[19];
    const float* Whh  = (const float*)d_in[20];
    const float* bih  = (const float*)d_in[21];
    const float* bhh  = (const float*)d_in[22];
    const float* Wout = (const float*)d_in[23];
    const float* bout = (const float*)d_in[24];

    float* out   = (float*)d_out;          // [probs(2), h1,c1, ..., h7,c7]
    float* z     = (float*)d_ws;           // 7 * 8192 floats
    float* parts = z + (size_t)7 * G4;     // 4 * 8192 floats
    float* ho[7]; float* co[7];
    for (int k = 0; k < 7; ++k) {
        ho[k] = out + 2 + (size_t)k * 2 * HH;
        co[k] = ho[k] + HH;
    }

    HPtrs hp; for (int i = 0; i < 7; ++i) hp.p[i] = hin[i];

    // Phase 1: all seven recurrent GEMVs in parallel (448 MB streamed once)
    gemv_all_hh<<<dim3(64, 7), 256, 0, stream>>>(Whh1, Whh, bih1, bhh1, bih, bhh,
                                                 Wih1, x, hp, z);
    // Layer 1 cell (gates fully known)
    lstm_cell<<<8, 256, 0, stream>>>(z, nullptr, cin[0], ho[0], co[0]);

    // Layers 2..7: sequential Wih GEMV (split-K x4) + cell
    for (int k = 1; k < 7; ++k) {
        const float* xa = ho[k - 1];
        const float* xb = (k >= 2) ? ho[k - 2] : nullptr;   // residual input
        const float* Wk = Wih + (size_t)(k - 1) * G4 * HH;
        gemv_ih_split<<<dim3(64, NSPL), 256, 0, stream>>>(Wk, xa, xb, parts);
        lstm_cell<<<8, 256, 0, stream>>>(z + (size_t)k * G4, parts, cin[k], ho[k], co[k]);
    }

    proj_softmax<<<1, 256, 0, stream>>>(Wout, bout, ho[6], out);
}